// DiceLoss_61529701483183
// MI455X (gfx1250) — compile-verified
//
#include <hip/hip_runtime.h>
#include <hip/hip_bf16.h>
#include <stdint.h>

// ---------------- problem constants ----------------
#define NCLS   19
#define NIMG   8
#define HW     262144            // 512*512
#define TILE_P 512               // pixels per TDM tile
#define BLK    128               // 4 wave32s, 4 pixels per thread
#define NWAVE  (BLK / 32)
#define HPAD   96                // 3 histograms x 32 padded bins per wave
#define CHUNKS (HW / TILE_P)     // 512 tiles per image
#define NCNT   (3 * NCLS)        // 57 global counters

#ifndef __has_builtin
#define __has_builtin(x) 0
#endif

#if defined(__AMDGCN__) && __has_builtin(__builtin_amdgcn_tensor_load_to_lds)
#define USE_TDM 1
#else
#define USE_TDM 0
#endif

#if defined(__has_include)
#if __has_include(<hip/amd_detail/amd_gfx1250_TDM.h>)
#define TDM_ARGS6 1
#endif
#endif
#ifndef TDM_ARGS6
#define TDM_ARGS6 0
#endif

#if USE_TDM
typedef unsigned int u32x4 __attribute__((ext_vector_type(4)));
typedef int          i32x4 __attribute__((ext_vector_type(4)));
typedef int          i32x8 __attribute__((ext_vector_type(8)));

// One TDM descriptor: load a [NCLS x TILE_P] fp32 tile (row stride = HW
// elements) from global into LDS.  D# packing per CDNA5 ISA ch.8.3/8.4.
__device__ __forceinline__ void tdm_load_tile(const float* gaddr,
                                              uint32_t lds_byte_addr) {
  const uint64_t ga = (uint64_t)(uintptr_t)gaddr;
  u32x4 g0;
  g0.x = 1u;                                           // count=1 (valid user D#)
  g0.y = lds_byte_addr;                                // lds_addr [63:32]
  g0.z = (uint32_t)ga;                                 // global_addr[31:0]
  g0.w = (uint32_t)((ga >> 32) & 0x1FFFFFFu)           // global_addr[56:32]
       | (2u << 30);                                   // type=2 ("image")
  i32x8 g1;
  g1[0] = (int)(2u << 16);                             // data_size=2 (4 bytes)
  g1[1] = 0;                                           // barrier addr=0, dim0 lo16=0
  g1[2] = (int)(((unsigned)NCLS << 16) | 0x4u);        // dim0 hi16 (HW>>16), dim1 lo16=19
  g1[3] = (int)((unsigned)TILE_P << 16);               // dim1 hi16=0, tile_dim0=512
  g1[4] = NCLS;                                        // tile_dim1=19, tile_dim2=0
  g1[5] = (int)(unsigned)HW;                           // tensor_dim0_stride[31:0]
  g1[6] = 0;
  g1[7] = 0;
  i32x4 z4 = {0, 0, 0, 0};
#if TDM_ARGS6
  i32x8 z8 = {0, 0, 0, 0, 0, 0, 0, 0};
  __builtin_amdgcn_tensor_load_to_lds(g0, g1, z4, z4, z8, 0);
#else
  __builtin_amdgcn_tensor_load_to_lds(g0, g1, z4, z4, 0);
#endif
}
#endif  // USE_TDM

__global__ __launch_bounds__(BLK) void dice_stage1(
    const float* __restrict__ pred, const long long* __restrict__ target,
    unsigned int* __restrict__ gcnt) {
  __shared__ float tile[NCLS * TILE_P];        // 38 KB staging tile
  __shared__ unsigned int hist[NWAVE * HPAD];  // per-wave padded histograms

  const int tid  = threadIdx.x;
  const int wid  = tid >> 5;
  const int n    = blockIdx.x / CHUNKS;
  const int base = (blockIdx.x % CHUNKS) * TILE_P;
  const float* gp = pred + (size_t)n * NCLS * HW + base;

  if (tid < HPAD) {
#pragma unroll
    for (int w = 0; w < NWAVE; ++w) hist[w * HPAD + tid] = 0u;
  }

#if USE_TDM
  if (tid < 32) {  // wave 0 issues the tensor DMA once; EXEC ignored by TDM
    tdm_load_tile(gp, (uint32_t)(uintptr_t)(void*)&tile[0]);
  }
#endif

  // Independent of the LDS tile: fetch the 4 target labels now (2x b128),
  // overlapping their latency with the tensor DMA.
  const int p0 = tid * 4;
  const long long* tg = target + (size_t)n * HW + base + p0;
  const int4 ta = *(const int4*)(tg);       // int64 low words at .x, .z
  const int4 tb = *(const int4*)(tg + 2);
  const int t0 = ta.x, t1 = ta.z, t2 = tb.x, t3 = tb.z;

#if USE_TDM
  if (tid < 32) __builtin_amdgcn_s_wait_tensorcnt(0);
  __syncthreads();
#else
  // Fallback: cooperative float4 loads, fully coalesced per channel row.
  for (int c = 0; c < NCLS; ++c) {
    const float4 v = *(const float4*)(gp + (size_t)c * HW + tid * 4);
    *(float4*)&tile[c * TILE_P + tid * 4] = v;
  }
  __syncthreads();
#endif

  // Branchless argmax over 19 channels for 4 pixels (ds_load_b128 per channel;
  // lane-consecutive 16B addresses -> conflict-free across 64 banks).
  float4 best = *(const float4*)&tile[p0];
  int i0 = 0, i1 = 0, i2 = 0, i3 = 0;
#pragma unroll
  for (int c = 1; c < NCLS; ++c) {
    const float4 v = *(const float4*)&tile[c * TILE_P + p0];
    if (v.x > best.x) { best.x = v.x; i0 = c; }
    if (v.y > best.y) { best.y = v.y; i1 = c; }
    if (v.z > best.z) { best.z = v.z; i2 = c; }
    if (v.w > best.w) { best.w = v.w; i3 = c; }
  }

  unsigned int* hp = hist + wid * HPAD;        // per-wave copies: no inter-wave
  unsigned int* ht = hp + 32;                  // contention on the 19 bins
  unsigned int* ho = hp + 64;
  atomicAdd(&hp[i0], 1u);  atomicAdd(&hp[i1], 1u);
  atomicAdd(&hp[i2], 1u);  atomicAdd(&hp[i3], 1u);
  atomicAdd(&ht[t0], 1u);  atomicAdd(&ht[t1], 1u);
  atomicAdd(&ht[t2], 1u);  atomicAdd(&ht[t3], 1u);
  if (i0 == t0) atomicAdd(&ho[t0], 1u);
  if (i1 == t1) atomicAdd(&ho[t1], 1u);
  if (i2 == t2) atomicAdd(&ho[t2], 1u);
  if (i3 == t3) atomicAdd(&ho[t3], 1u);

  __syncthreads();
  if (tid < NCNT) {
    const int grp = tid / NCLS, idx = tid % NCLS;
    unsigned int s = 0u;
#pragma unroll
    for (int w = 0; w < NWAVE; ++w) s += hist[w * HPAD + grp * 32 + idx];
    atomicAdd(&gcnt[tid], s);
  }
}

__global__ void dice_stage2(const unsigned int* __restrict__ gcnt,
                            float* __restrict__ out) {
  if (threadIdx.x == 0 && blockIdx.x == 0) {
    float s = 0.0f;
    for (int c = 0; c < NCLS; ++c) {
      const float p = (float)gcnt[c];
      const float t = (float)gcnt[NCLS + c];
      const float o = (float)gcnt[2 * NCLS + c];
      s += 2.0f * o / (p + t + 1.0f);
    }
    out[0] = 1.0f - s / (float)(NIMG * NCLS);
  }
}

extern "C" void kernel_launch(void* const* d_in, const int* in_sizes, int n_in,
                              void* d_out, int out_size, void* d_ws, size_t ws_size,
                              hipStream_t stream) {
  const float* pred       = (const float*)d_in[0];
  const long long* target = (const long long*)d_in[1];
  unsigned int* cnt       = (unsigned int*)d_ws;
  float* out              = (float*)d_out;

  hipMemsetAsync(cnt, 0, NCNT * sizeof(unsigned int), stream);
  dice_stage1<<<NIMG * CHUNKS, BLK, 0, stream>>>(pred, target, cnt);
  dice_stage2<<<1, 32, 0, stream>>>(cnt, out);
}